// PercolationM_66048007078107
// MI455X (gfx1250) — compile-verified
//
#include <hip/hip_runtime.h>
#include <hip/hip_bf16.h>
#include <stdint.h>

#define NPB   262144          // elements per batch element (256*32*32)
#define SPLIT 8               // workgroups per batch element
#define CHUNK (NPB / SPLIT)   // 32768 elements per WG
#define TPB   256             // threads per WG (8 wave32)
#define BINS  256
#define QD    4               // async staging ring depth (power of 2)
#define STAGE_BYTES (TPB * 16)          // 4KB per stage buffer
#define ITERS (CHUNK / (TPB * 4))       // 32 stages per WG

// ---------------------------------------------------------------------------
// Zero the per-batch global histograms in workspace.
// ---------------------------------------------------------------------------
__global__ void __launch_bounds__(TPB) zero_kernel(unsigned* __restrict__ gh) {
    gh[(size_t)blockIdx.x * BINS + threadIdx.x] = 0u;
}

// ---------------------------------------------------------------------------
// Streaming histogram kernel. Each WG handles a contiguous 32768-element chunk
// of one batch element. Data is streamed global->LDS with CDNA5 async copies
// (4-deep ring, ASYNCcnt-tracked), then binned into per-wave LDS
// sub-histograms with ds_add_u32 atomics. Per-WG partials are atomically
// accumulated into the per-batch global histogram.
// ---------------------------------------------------------------------------
__global__ void __launch_bounds__(TPB) hist_kernel(const int* __restrict__ in,
                                                   unsigned* __restrict__ ghist) {
    __shared__ __align__(16) unsigned stage[QD * TPB * 4];  // 4 x 4KB staging ring
    __shared__ unsigned hist[8 * BINS];                     // per-wave histograms

    const unsigned tid = threadIdx.x;
    const unsigned wid = tid >> 5;                          // wave32 id within WG

    for (unsigned t = tid; t < 8 * BINS; t += TPB) hist[t] = 0u;
    __syncthreads();

    const unsigned batch = blockIdx.x / SPLIT;
    const unsigned chunk = blockIdx.x % SPLIT;

    // Per-thread byte offset of the first 16B packet (fits in 32-bit: <134MB).
    const unsigned gbyte0 = (batch * (unsigned)NPB + chunk * (unsigned)CHUNK + tid * 4u) * 4u;
    // Workgroup-relative LDS byte address of this thread's slot in stage 0.
    const unsigned lds0 = (unsigned)(uintptr_t)(&stage[0]) + tid * 16u;
    const unsigned long long sbase = (unsigned long long)(uintptr_t)in;

    // Issue one async stage: 32 lanes x 16B = 512B per wave per issue.
#define ISSUE(s)                                                               \
    do {                                                                       \
        const unsigned _v = gbyte0 + (unsigned)(s) * (unsigned)STAGE_BYTES;    \
        const unsigned _l = lds0 + ((unsigned)(s) & (QD - 1)) * (unsigned)STAGE_BYTES; \
        asm volatile("global_load_async_to_lds_b128 %0, %1, %2"                \
                     :: "v"(_l), "v"(_v), "s"(sbase) : "memory");              \
    } while (0)

    // Prologue: fill the ring minus one.
    ISSUE(0);
    ISSUE(1);
    ISSUE(2);

    for (int i = 0; i < ITERS; ++i) {
        if (i + QD - 1 < ITERS) {
            ISSUE(i + QD - 1);
            // Async loads retire in order: asynccnt<=3 -> stage i is resident.
            asm volatile("s_wait_asynccnt 0x3" ::: "memory");
        } else if (i + 2 < ITERS) {
            asm volatile("s_wait_asynccnt 0x2" ::: "memory");
        } else if (i + 1 < ITERS) {
            asm volatile("s_wait_asynccnt 0x1" ::: "memory");
        } else {
            asm volatile("s_wait_asynccnt 0x0" ::: "memory");
        }

        const unsigned* sp = &stage[(unsigned)(i & (QD - 1)) * (TPB * 4u) + tid * 4u];
        const uint4 v = *reinterpret_cast<const uint4*>(sp);

        unsigned* h = &hist[wid * BINS];
        atomicAdd(&h[v.x & 255u], 1u);
        atomicAdd(&h[v.y & 255u], 1u);
        atomicAdd(&h[v.z & 255u], 1u);
        atomicAdd(&h[v.w & 255u], 1u);
    }
#undef ISSUE
    __syncthreads();

    // tid == bin (TPB == BINS): fold the 8 wave-private histograms, push out.
    unsigned s = 0;
#pragma unroll
    for (int w = 0; w < 8; ++w) s += hist[w * BINS + tid];
    atomicAdd(&ghist[(size_t)batch * BINS + tid], s);
}

// ---------------------------------------------------------------------------
// Per-batch max over 256 bins -> float.
// ---------------------------------------------------------------------------
__global__ void __launch_bounds__(BINS) max_kernel(const unsigned* __restrict__ gh,
                                                   float* __restrict__ out) {
    __shared__ unsigned red[BINS];
    const unsigned tid = threadIdx.x;
    red[tid] = gh[(size_t)blockIdx.x * BINS + tid];
    __syncthreads();
    for (int st = BINS / 2; st > 0; st >>= 1) {
        if ((int)tid < st) {
            const unsigned a = red[tid], b = red[tid + st];
            red[tid] = (a > b) ? a : b;
        }
        __syncthreads();
    }
    if (tid == 0) out[blockIdx.x] = (float)red[0];
}

// ---------------------------------------------------------------------------
// Fallback (no workspace needed): one WG of 1024 threads per batch element,
// 32 wave-private LDS histograms, inline max reduction.
// ---------------------------------------------------------------------------
__global__ void __launch_bounds__(1024) mono_kernel(const int* __restrict__ in,
                                                    float* __restrict__ out) {
    __shared__ unsigned hist[32 * BINS];   // 32 KB
    __shared__ unsigned binsum[BINS];

    const unsigned tid = threadIdx.x;
    const unsigned wid = tid >> 5;
    for (unsigned t = tid; t < 32 * BINS; t += 1024) hist[t] = 0u;
    __syncthreads();

    const uint4* p = reinterpret_cast<const uint4*>(in + (size_t)blockIdx.x * NPB);
    for (unsigned idx = tid; idx < NPB / 4; idx += 1024) {
        const uint4 v = p[idx];
        unsigned* h = &hist[wid * BINS];
        atomicAdd(&h[v.x & 255u], 1u);
        atomicAdd(&h[v.y & 255u], 1u);
        atomicAdd(&h[v.z & 255u], 1u);
        atomicAdd(&h[v.w & 255u], 1u);
    }
    __syncthreads();

    if (tid < BINS) {
        unsigned s = 0;
#pragma unroll
        for (int w = 0; w < 32; ++w) s += hist[w * BINS + tid];
        binsum[tid] = s;
    }
    __syncthreads();
    for (int st = BINS / 2; st > 0; st >>= 1) {
        if (tid < (unsigned)st) {
            const unsigned a = binsum[tid], b = binsum[tid + st];
            binsum[tid] = (a > b) ? a : b;
        }
        __syncthreads();
    }
    if (tid == 0) out[blockIdx.x] = (float)binsum[0];
}

// ---------------------------------------------------------------------------
extern "C" void kernel_launch(void* const* d_in, const int* in_sizes, int n_in,
                              void* d_out, int out_size, void* d_ws, size_t ws_size,
                              hipStream_t stream) {
    (void)n_in; (void)out_size;
    const int*  in  = (const int*)d_in[0];
    float*      out = (float*)d_out;
    const int   B   = in_sizes[0] / NPB;          // 128 batch elements

    unsigned* gh = (unsigned*)d_ws;
    if (ws_size >= (size_t)B * BINS * sizeof(unsigned)) {
        zero_kernel<<<B, TPB, 0, stream>>>(gh);
        hist_kernel<<<B * SPLIT, TPB, 0, stream>>>(in, gh);
        max_kernel<<<B, BINS, 0, stream>>>(gh, out);
    } else {
        mono_kernel<<<B, 1024, 0, stream>>>(in, out);
    }
}